// pDANSE_52218212385475
// MI455X (gfx1250) — compile-verified
//
#include <hip/hip_runtime.h>
#include <hip/hip_bf16.h>
#include <math.h>

// dims
#define L_ 200
#define N_ 32
#define T_ 500
#define O_ 10
#define S_ 10
#define HD_ 64

typedef __attribute__((ext_vector_type(2))) float v2f;
typedef __attribute__((ext_vector_type(8))) float v8f;

// ---------------------------------------------------------------------------
// Kernel 0: zero the quad accumulator (L*N floats)
// ---------------------------------------------------------------------------
__global__ void kInit(float* quad) {
    int i = blockIdx.x * blockDim.x + threadIdx.x;
    if (i < L_ * N_) quad[i] = 0.0f;
}

// ---------------------------------------------------------------------------
// Kernel A: per-batch Cholesky of Cw, logdet, Linv = L^-1, G = Linv@Hm.
// Stores Bpad[n][12][16] with Bpad[s][o] = G[o][s] (zero padded) for WMMA B.
// One thread per n; tiny workload, scalar is fine.
// ---------------------------------------------------------------------------
__global__ void kDecomp(const float* __restrict__ Cw, const float* __restrict__ Hm,
                        float* __restrict__ Bpad, float* __restrict__ Linv_out,
                        float* __restrict__ logdet) {
    int n = threadIdx.x;
    if (n >= N_) return;
    const float* C = Cw + n * O_ * O_;
    float Lc[O_][O_];
    float ld = 0.0f;
    for (int i = 0; i < O_; ++i) {
        for (int j = 0; j <= i; ++j) {
            float s = C[i * O_ + j];
            for (int k = 0; k < j; ++k) s -= Lc[i][k] * Lc[j][k];
            if (i == j) { Lc[i][i] = sqrtf(s); ld += logf(Lc[i][i]); }
            else        { Lc[i][j] = s / Lc[j][j]; }
        }
    }
    logdet[n] = 2.0f * ld;
    // Linv (lower triangular inverse) by forward substitution per column
    float Li[O_][O_];
    for (int j = 0; j < O_; ++j) {
        for (int i = 0; i < O_; ++i) {
            if (i < j) { Li[i][j] = 0.0f; continue; }
            float s = (i == j) ? 1.0f : 0.0f;
            for (int k = j; k < i; ++k) s -= Lc[i][k] * Li[k][j];
            Li[i][j] = s / Lc[i][i];
        }
    }
    for (int i = 0; i < O_; ++i)
        for (int j = 0; j < O_; ++j)
            Linv_out[n * O_ * O_ + i * O_ + j] = Li[i][j];
    // G[o][s] = sum_p Linv[o][p] * Hm[p][s];  Bpad[s][o] = G[o][s], padded 12x16
    float* B = Bpad + n * 12 * 16;
    for (int s = 0; s < 12; ++s) {
        for (int o = 0; o < 16; ++o) {
            float g = 0.0f;
            if (s < S_ && o < O_)
                for (int p = 0; p < O_; ++p) g += Li[o][p] * Hm[p * S_ + s];
            B[s * 16 + o] = g;
        }
    }
}

// ---------------------------------------------------------------------------
// Kernel B0: x = Yi @ W + b  (N*T, 3Hd) input projections (61 MFLOP, VALU)
// ---------------------------------------------------------------------------
__global__ void kXProj(const float* __restrict__ Yi, const float* __restrict__ W,
                       const float* __restrict__ b, float* __restrict__ x) {
    int idx = blockIdx.x * blockDim.x + threadIdx.x;
    if (idx >= N_ * T_ * 3 * HD_) return;
    int c = idx % (3 * HD_);
    int nt = idx / (3 * HD_);
    const float* y = Yi + nt * O_;
    float s = b[c];
    #pragma unroll
    for (int o = 0; o < O_; ++o) s += y[o] * W[o * (3 * HD_) + c];
    x[idx] = s;
}

// ---------------------------------------------------------------------------
// Kernel B1: GRU scan over T. One workgroup, 8 waves. Per step h@U is a
// 32x64x192 fp32 GEMM done with V_WMMA_F32_16X16X4_F32 (24 tiles, 3/wave),
// U held in registers, h/g staged in LDS. Stores h to global (N,T,Hd).
// ---------------------------------------------------------------------------
__launch_bounds__(256, 1)
__global__ void kGRU(const float* __restrict__ x, const float* __restrict__ U,
                     float* __restrict__ hws) {
    __shared__ float hS[N_ * HD_];          // 32x64
    __shared__ float gS[N_ * 3 * HD_];      // 32x192
    const int tid = threadIdx.x;
    const int wave = tid >> 5;
    const int lane = tid & 31;
    const int laneLo = lane & 15;
    const int kOff = (lane >> 4) * 2;

    for (int i = tid; i < N_ * HD_; i += 256) hS[i] = 0.0f;

    // U tiles pinned in registers: wave handles flattened tiles wave*3..+2
    // tile tt -> (mi = tt/12 M-tile of n, ni = tt%12 N-tile of 192 cols)
    v2f Breg[3][16];
    #pragma unroll
    for (int j = 0; j < 3; ++j) {
        int tt = wave * 3 + j;
        int ni = tt % 12;
        int col = ni * 16 + laneLo;
        #pragma unroll
        for (int kk = 0; kk < 16; ++kk) {
            Breg[j][kk].x = U[(kk * 4 + kOff) * (3 * HD_) + col];
            Breg[j][kk].y = U[(kk * 4 + kOff + 1) * (3 * HD_) + col];
        }
    }
    __syncthreads();

    for (int t = 0; t < T_; ++t) {
        // ---- g = h @ U via WMMA ----
        #pragma unroll
        for (int j = 0; j < 3; ++j) {
            int tt = wave * 3 + j;
            int mi = tt / 12;
            int ni = tt % 12;
            int row = mi * 16 + laneLo;      // n index of A row
            v8f c = {};
            #pragma unroll
            for (int kk = 0; kk < 16; ++kk) {
                v2f a;
                a.x = hS[row * HD_ + kk * 4 + kOff];
                a.y = hS[row * HD_ + kk * 4 + kOff + 1];
                c = __builtin_amdgcn_wmma_f32_16x16x4_f32(
                        false, a, false, Breg[j][kk], (short)0, c, false, false);
            }
            int colD = ni * 16 + laneLo;
            int rbase = mi * 16 + ((lane & 16) ? 8 : 0);
            #pragma unroll
            for (int jj = 0; jj < 8; ++jj)
                gS[(rbase + jj) * (3 * HD_) + colD] = c[jj];
        }
        __syncthreads();
        // ---- gate combine: 8 (n,hd) elements per thread ----
        #pragma unroll
        for (int e = 0; e < 8; ++e) {
            int idx = tid * 8 + e;
            int n = idx >> 6;
            int hd = idx & 63;
            const float* xb = x + ((size_t)(n * T_ + t)) * (3 * HD_);
            float xz = xb[hd];
            float xr = xb[HD_ + hd];
            float xn = xb[2 * HD_ + hd];
            float hz = gS[n * (3 * HD_) + hd];
            float hr = gS[n * (3 * HD_) + HD_ + hd];
            float hn = gS[n * (3 * HD_) + 2 * HD_ + hd];
            float z = 1.0f / (1.0f + __expf(-(xz + hz)));
            float r = 1.0f / (1.0f + __expf(-(xr + hr)));
            float ng = tanhf(xn + r * hn);
            float hold = hS[idx];
            float hnew = (1.0f - z) * ng + z * hold;
            hS[idx] = hnew;
            hws[((size_t)(n * T_ + t)) * HD_ + hd] = hnew;
        }
        __syncthreads();
    }
}

// ---------------------------------------------------------------------------
// Kernel C: per (n,t): mu = h@Wmu+bmu, sd = sqrt(softplus(h@Wv+bv)+1e-3),
// d0 = y - Hm@mu, u = Linv@d0. One wave per (n,t), wave32 shuffles.
// Stores u and sd zero-padded to 16 for the WMMA kernel.
// ---------------------------------------------------------------------------
__global__ void kProj(const float* __restrict__ hws, const float* __restrict__ Yi,
                      const float* __restrict__ Wmu, const float* __restrict__ bmu,
                      const float* __restrict__ Wv, const float* __restrict__ bv,
                      const float* __restrict__ Hm, const float* __restrict__ Linv,
                      float* __restrict__ uws, float* __restrict__ sdws) {
    int w = blockIdx.x * (blockDim.x >> 5) + (threadIdx.x >> 5);
    int lane = threadIdx.x & 31;
    if (w >= N_ * T_) return;
    int n = w / T_;
    const float* h = hws + (size_t)w * HD_;

    float proj = 0.0f;            // lanes 0-9: mu_s ; lanes 10-19: raw v_s
    if (lane < S_) {
        for (int k = 0; k < HD_; ++k) proj += h[k] * Wmu[k * S_ + lane];
        proj += bmu[lane];
    } else if (lane < 2 * S_) {
        int s = lane - S_;
        for (int k = 0; k < HD_; ++k) proj += h[k] * Wv[k * S_ + s];
        proj += bv[s];
    }
    float sd = 0.0f;
    if (lane >= S_ && lane < 2 * S_) {
        float vr = proj;
        float sp = (vr > 20.0f) ? vr : log1pf(__expf(vr));
        sd = sqrtf(sp + 1e-3f);
    }
    // d0_o = y_o - sum_s Hm[o][s]*mu_s
    float d0acc = 0.0f;
    for (int ss = 0; ss < S_; ++ss) {
        float mus = __shfl(proj, ss, 32);
        if (lane < O_) d0acc += Hm[lane * S_ + ss] * mus;
    }
    float d0v = 0.0f;
    if (lane < O_) d0v = Yi[(size_t)w * O_ + lane] - d0acc;
    // u_o = sum_{p<=o} Linv[o][p] * d0_p
    float uacc = 0.0f;
    for (int p = 0; p < O_; ++p) {
        float dp = __shfl(d0v, p, 32);
        if (lane < O_ && p <= lane) uacc += Linv[n * O_ * O_ + lane * O_ + p] * dp;
    }
    if (lane < 16) uws[(size_t)w * 16 + lane] = (lane < O_) ? uacc : 0.0f;
    // broadcast sd (held in lanes 10..19) to its s-slot; lanes>=16 store
    float sdbr = __shfl(sd, (lane & 15) + S_, 32);
    if (lane >= 16) sdws[(size_t)w * 16 + (lane & 15)] = ((lane & 15) < S_) ? sdbr : 0.0f;
}

// ---------------------------------------------------------------------------
// Kernel D: the 128 MB eps sweep. Block = (n, 16-t chunk); 8 waves loop over
// particles l. Each tile: A = (sd ⊙ eps) 16 rows (t) x 12 K (s, padded),
// B = G_n^T padded 12x16 -> 3x V_WMMA_F32_16X16X4_F32; epilogue accumulates
// ||u - w||^2 into quad[l][n]. eps read exactly once with non-temporal loads
// (TH=NT) so the single-use 128 MB stream does not evict the 200x-reused
// u/sd/B working set from the 192 MB L2.
// ---------------------------------------------------------------------------
__launch_bounds__(256, 1)
__global__ void kQuad(const float* __restrict__ eps, const float* __restrict__ Bpad,
                      const float* __restrict__ uws, const float* __restrict__ sdws,
                      float* __restrict__ quad) {
    __shared__ float sdS[16][16];
    __shared__ float uS[16][16];
    const int n = blockIdx.x & 31;
    const int tc = blockIdx.x >> 5;      // 0..31
    const int t0 = tc * 16;
    const int tid = threadIdx.x;
    const int wave = tid >> 5;
    const int lane = tid & 31;
    const int laneLo = lane & 15;
    const int kOff = (lane >> 4) * 2;

    // stage sd / u for this (n, t-chunk); invalid t -> 0 (masks padded rows)
    {
        int r = tid >> 4, c = tid & 15;
        int t = t0 + r;
        sdS[r][c] = (t < T_) ? sdws[((size_t)n * T_ + t) * 16 + c] : 0.0f;
        uS[r][c]  = (t < T_) ? uws[((size_t)n * T_ + t) * 16 + c] : 0.0f;
    }
    // B = G_n^T (padded) in registers
    v2f Breg[3];
    {
        const float* B = Bpad + n * 12 * 16;
        #pragma unroll
        for (int kk = 0; kk < 3; ++kk) {
            Breg[kk].x = B[(kk * 4 + kOff) * 16 + laneLo];
            Breg[kk].y = B[(kk * 4 + kOff + 1) * 16 + laneLo];
        }
    }
    __syncthreads();

    const int row = laneLo;              // t-row inside chunk
    const int tRow = t0 + row;
    const bool tValid = tRow < T_;
    float sA[3][2];
    #pragma unroll
    for (int kk = 0; kk < 3; ++kk) {
        int s0 = kk * 4 + kOff;
        sA[kk][0] = (s0 < S_) ? sdS[row][s0] : 0.0f;
        sA[kk][1] = (s0 + 1 < S_) ? sdS[row][s0 + 1] : 0.0f;
    }

    for (int l = wave; l < L_; l += 8) {
        const float* erow = eps +
            ((((size_t)l * N_ + n) * T_ + (tValid ? tRow : 0))) * S_;
        if (l + 8 < L_) {
            const float* enext = eps +
                ((((size_t)(l + 8) * N_ + n) * T_ + (tValid ? tRow : 0))) * S_;
            __builtin_prefetch(enext, 0, 1);   // global_prefetch_b8
        }
        v8f c = {};
        #pragma unroll
        for (int kk = 0; kk < 3; ++kk) {
            int s0 = kk * 4 + kOff;
            v2f e = {0.0f, 0.0f};
            if (tValid && s0 < S_)           // s0 even & row base 40B -> 8B aligned
                e = __builtin_nontemporal_load((const v2f*)(erow + s0));
            v2f a;
            a.x = e.x * sA[kk][0];
            a.y = e.y * sA[kk][1];
            c = __builtin_amdgcn_wmma_f32_16x16x4_f32(
                    false, a, false, Breg[kk], (short)0, c, false, false);
        }
        // epilogue: quad contribution = sum over (t-row, o<10) of (u - w)^2
        float acc = 0.0f;
        if (laneLo < O_) {
            int rbase = (lane & 16) ? 8 : 0;
            #pragma unroll
            for (int jj = 0; jj < 8; ++jj) {
                float r_ = uS[rbase + jj][laneLo] - c[jj];
                acc += r_ * r_;
            }
        }
        #pragma unroll
        for (int m = 16; m >= 1; m >>= 1) acc += __shfl_xor(acc, m, 32);
        if (lane == 0) atomicAdd(&quad[l * N_ + n], acc);
    }
}

// ---------------------------------------------------------------------------
// Kernel E: final log-prob assembly + reduction to scalar ELBO.
// ---------------------------------------------------------------------------
__global__ void kFinal(const float* __restrict__ quad, const float* __restrict__ logdet,
                       float* __restrict__ out) {
    __shared__ float red[256];
    const float LOG2PI = 1.8378770664093455f;
    int tid = threadIdx.x;
    float s = 0.0f;
    for (int i = tid; i < L_ * N_; i += 256) {
        int n = i & 31;
        float lp = -0.5f * O_ * T_ * LOG2PI - 0.5f * T_ * logdet[n] - 0.5f * quad[i];
        s += lp;
    }
    red[tid] = s;
    __syncthreads();
    for (int st = 128; st > 0; st >>= 1) {
        if (tid < st) red[tid] += red[tid + st];
        __syncthreads();
    }
    if (tid == 0) out[0] = red[0] / ((float)L_ * N_ * T_ * O_);
}

// ---------------------------------------------------------------------------
extern "C" void kernel_launch(void* const* d_in, const int* in_sizes, int n_in,
                              void* d_out, int out_size, void* d_ws, size_t ws_size,
                              hipStream_t stream) {
    const float* Yi  = (const float*)d_in[0];
    const float* Cw  = (const float*)d_in[1];
    const float* eps = (const float*)d_in[2];
    const float* W   = (const float*)d_in[3];
    const float* U   = (const float*)d_in[4];
    const float* b   = (const float*)d_in[5];
    const float* Wmu = (const float*)d_in[6];
    const float* bmu = (const float*)d_in[7];
    const float* Wv  = (const float*)d_in[8];
    const float* bv  = (const float*)d_in[9];
    const float* Hm  = (const float*)d_in[10];

    float* ws     = (float*)d_ws;
    float* Bpad   = ws;                       // 32*12*16      = 6144
    float* Linv   = Bpad + N_ * 12 * 16;      // 32*100        = 3200
    float* logdet = Linv + N_ * O_ * O_;      // 32
    float* quad   = logdet + N_;              // 6400
    float* x      = quad + L_ * N_;           // 32*500*192    = 3,072,000
    float* hws    = x + (size_t)N_ * T_ * 3 * HD_;  // 1,024,000
    float* uws    = hws + (size_t)N_ * T_ * HD_;    // 256,000
    float* sdws   = uws + (size_t)N_ * T_ * 16;     // 256,000
    // total ~18.5 MB of workspace

    kInit<<<(L_ * N_ + 255) / 256, 256, 0, stream>>>(quad);
    kDecomp<<<1, 32, 0, stream>>>(Cw, Hm, Bpad, Linv, logdet);
    kXProj<<<(N_ * T_ * 3 * HD_ + 255) / 256, 256, 0, stream>>>(Yi, W, b, x);
    kGRU<<<1, 256, 0, stream>>>(x, U, hws);
    kProj<<<(N_ * T_ + 7) / 8, 256, 0, stream>>>(hws, Yi, Wmu, bmu, Wv, bv, Hm, Linv,
                                                 uws, sdws);
    kQuad<<<N_ * 32, 256, 0, stream>>>(eps, Bpad, uws, sdws, quad);
    kFinal<<<1, 256, 0, stream>>>(quad, logdet, (float*)d_out);
}